// BillehColumn_20830591386291
// MI455X (gfx1250) — compile-verified
//
#include <hip/hip_runtime.h>
#include <stdint.h>

// Problem constants (from reference)
#define TT 20
#define BB 4
#define NN 50000
#define RR 4
#define EE 1600000
#define DD 5
#define NRR (NN * RR)          // 200000
#define NWRD ((NN + 31) / 32)  // 1563 32-bit words of spike bits per (slot,batch)
#define ZB_U32 (DD * NWRD * 4) // total u32 words in the bit ring (slot-major, then word, then batch)

#define EDGE_TILE 512                               // edges per TDM tile
#define N_TILES ((EE + EDGE_TILE - 1) / EDGE_TILE)  // 3125
#define EDGE_BLOCKS 1563                            // ~2 tiles per one-wave block (pipelined)

typedef unsigned int u32x4 __attribute__((ext_vector_type(4)));
typedef int          i32x8 __attribute__((ext_vector_type(8)));
typedef int          i32x4 __attribute__((ext_vector_type(4)));

// ---------------------------------------------------------------------------
// Tensor Data Mover: 1-D tile (32-bit elements) global -> LDS.
// D# packing per CDNA5 ISA ch.7/8 (group0: count/lds_addr/global_addr/type,
// group1: data_size, tensor_dim0 (OOB zero-fill bound), tile_dim0).
// Issued once per wave (EXEC-independent DMA); completion via TENSORcnt,
// in-order per wave -> enables double-buffered pipelining below.
// ---------------------------------------------------------------------------
__device__ __forceinline__ void tdm_load_1d_dwords(uint32_t lds_byte_addr,
                                                   const void* gptr,
                                                   uint32_t tile_elems,
                                                   uint32_t tensor_elems_remaining) {
  uint64_t ga = (uint64_t)(uintptr_t)gptr;
  u32x4 g0;
  g0[0] = 1u;                                   // count=1 (valid user descriptor)
  g0[1] = lds_byte_addr;                        // LDS destination (byte address)
  g0[2] = (uint32_t)ga;                         // global_addr[31:0]
  g0[3] = (uint32_t)((ga >> 32) & 0x1FFFFFFu)   // global_addr[56:32]
        | (2u << 30);                           // type = 2 ("image")

  uint32_t td0 = tensor_elems_remaining;        // OOB bound (reads past -> 0)
  uint32_t td1 = 1u;
  i32x8 g1;
  g1[0] = (int)(2u << 16);                      // wg_mask=0, data_size=2 (4 bytes)
  g1[1] = (int)((td0 & 0xFFFFu) << 16);         // tensor_dim0[15:0]
  g1[2] = (int)((td0 >> 16) | ((td1 & 0xFFFFu) << 16)); // td0[31:16] | td1[15:0]
  g1[3] = (int)((td1 >> 16) | ((tile_elems & 0xFFFFu) << 16)); // tile_dim0
  g1[4] = 0;                                    // tile_dim1=0, tile_dim2=0 (1-D)
  g1[5] = (int)td0;                             // tensor_dim0_stride (unused, 1-D)
  g1[6] = 0;
  g1[7] = 0;
  i32x4 gz4 = {0, 0, 0, 0};                     // groups 2/3 unused (<=2-D tensor)
  i32x8 gz8 = {0, 0, 0, 0, 0, 0, 0, 0};         // extra group (unused)
  __builtin_amdgcn_tensor_load_to_lds(g0, g1, gz4, gz4, gz8, 0);
}

// ---------------------------------------------------------------------------
// Edge kernel (defined FIRST so the disasm snippet shows the TDM sequence).
// One wave (32 threads) per block; grid-stride over 512-edge tiles with a
// double-buffered TDM pipeline: next tile's rows/cols/weights DMA while the
// current tile is processed from LDS.
//   - s_wait_dscnt 0 guards LDS buffer reuse vs the wave's in-flight ds reads
//   - s_wait_tensorcnt 3 releases the current tile while the next stays in flight
//   - spike gather: one b128 load of 4 batch words from the bit-packed ring;
//     whole edge skipped when no batch spiked
//   - scatter: global_atomic_add_f32 into the L2-resident rec accumulator
// rec[b][row] += w[e] * z_ring[(t-1-d) mod D][b][n],  d = col / N, n = col % N
// ---------------------------------------------------------------------------
__global__ void __launch_bounds__(32)
glif_edges(const int* __restrict__ rows, const int* __restrict__ cols,
           const float* __restrict__ weights,
           const uint32_t* __restrict__ zbits,
           float* __restrict__ rec, int t) {
  __shared__ uint32_t lr[2][EDGE_TILE];
  __shared__ uint32_t lc[2][EDGE_TILE];
  __shared__ float    lw[2][EDGE_TILE];

  const uint4* zb4 = (const uint4*)zbits;  // [slot*NWRD + word] -> 4 batch words
  int lane = threadIdx.x;

  int tile = blockIdx.x;
  if (tile >= N_TILES) return;
  int buf = 0;

  {
    int base = tile * EDGE_TILE;
    int rem = EE - base;
    int cnt = rem < EDGE_TILE ? rem : EDGE_TILE;
    tdm_load_1d_dwords((uint32_t)(uintptr_t)&lr[0][0], rows + base,    (uint32_t)cnt, (uint32_t)rem);
    tdm_load_1d_dwords((uint32_t)(uintptr_t)&lc[0][0], cols + base,    (uint32_t)cnt, (uint32_t)rem);
    tdm_load_1d_dwords((uint32_t)(uintptr_t)&lw[0][0], weights + base, (uint32_t)cnt, (uint32_t)rem);
  }

  while (tile < N_TILES) {
    int next = tile + (int)gridDim.x;
    if (next < N_TILES) {
      // Stage next tile into the other buffer while this one is processed.
      int nb = next * EDGE_TILE;
      int nrem = EE - nb;
      int ncnt = nrem < EDGE_TILE ? nrem : EDGE_TILE;
      asm volatile("s_wait_dscnt 0x0" ::: "memory");  // buffer-reuse guard
      tdm_load_1d_dwords((uint32_t)(uintptr_t)&lr[buf ^ 1][0], rows + nb,    (uint32_t)ncnt, (uint32_t)nrem);
      tdm_load_1d_dwords((uint32_t)(uintptr_t)&lc[buf ^ 1][0], cols + nb,    (uint32_t)ncnt, (uint32_t)nrem);
      tdm_load_1d_dwords((uint32_t)(uintptr_t)&lw[buf ^ 1][0], weights + nb, (uint32_t)ncnt, (uint32_t)nrem);
      __builtin_amdgcn_s_wait_tensorcnt(3);  // current tile landed; next in flight
    } else {
      __builtin_amdgcn_s_wait_tensorcnt(0);
    }

    int base = tile * EDGE_TILE;
    int rem = EE - base;
    int cnt = rem < EDGE_TILE ? rem : EDGE_TILE;

    for (int i = lane; i < cnt; i += 32) {
      uint32_t row = lr[buf][i];
      uint32_t col = lc[buf][i];
      float    wt  = lw[buf][i];
      uint32_t d = col / (uint32_t)NN;       // delay slot index in z_buf layout
      uint32_t n = col - d * (uint32_t)NN;   // source neuron
      uint32_t slot = (uint32_t)(t + DD - 1 - (int)d) % DD;  // ring slot of z(t-1-d)
      uint4 zw = zb4[slot * NWRD + (n >> 5)];
      uint32_t bit = n & 31u;
      if (((zw.x | zw.y | zw.z | zw.w) >> bit) & 1u) {
        if ((zw.x >> bit) & 1u) atomicAdd(&rec[0 * NRR + row], wt);
        if ((zw.y >> bit) & 1u) atomicAdd(&rec[1 * NRR + row], wt);
        if ((zw.z >> bit) & 1u) atomicAdd(&rec[2 * NRR + row], wt);
        if ((zw.w >> bit) & 1u) atomicAdd(&rec[3 * NRR + row], wt);
      }
    }
    buf ^= 1;
    tile = next;
  }
}

// ---------------------------------------------------------------------------
// Neuron kernel: one thread per (b, n). Fuses the full GLIF3 state update.
// All R=4 groups are 16B aligned -> b128 vector traffic for rec/x/psc/params.
// Consumes+zeroes rec, emits spike to d_out, and bit-packs z(t) into ring
// slot t%D with a wave32 ballot (one v_ballot_w32 + one u32 store per wave).
// ---------------------------------------------------------------------------
__global__ void glif_neurons(const float* __restrict__ x,
                             const float* __restrict__ syn_decay,
                             const float* __restrict__ psc_initial,
                             const float* __restrict__ asc_amps,
                             const float* __restrict__ exp_k1,
                             const float* __restrict__ exp_k2,
                             const float* __restrict__ decay,
                             const float* __restrict__ current_factor,
                             const float* __restrict__ v_th,
                             const float* __restrict__ e_l,
                             const float* __restrict__ v_reset,
                             const float* __restrict__ t_ref,
                             const float* __restrict__ param_g,
                             float* __restrict__ rec,
                             float* __restrict__ psc_rise, float* __restrict__ psc,
                             float* __restrict__ v, float* __restrict__ rref,
                             float* __restrict__ asc1, float* __restrict__ asc2,
                             uint32_t* __restrict__ zbits,
                             float* __restrict__ out, int t) {
  int n = blockIdx.x * blockDim.x + threadIdx.x;
  int b = blockIdx.y;
  int pslot = (t + DD - 1) % DD;   // holds z(t-1)
  int wslot = t % DD;              // will hold z(t)
  float z = 0.0f;

  if (n < NN) {
    int idx = b * NN + n;
    uint32_t pw = zbits[((uint32_t)pslot * NWRD + (n >> 5)) * 4 + b];
    float prev_z = (float)((pw >> (n & 31)) & 1u);

    int base4 = b * NRR + n * RR;
    int xbase = ((t * BB + b) * NN + n) * RR;

    float4 rv = *(float4*)&rec[base4];
    *(float4*)&rec[base4] = make_float4(0.f, 0.f, 0.f, 0.f);  // reset for step t+1
    float4 xx = *(const float4*)&x[xbase];
    float4 sd = *(const float4*)&syn_decay[n * RR];
    float4 pi = *(const float4*)&psc_initial[n * RR];
    float4 pr = *(float4*)&psc_rise[base4];
    float4 pc = *(float4*)&psc[base4];

    float4 inp = make_float4(rv.x + xx.x, rv.y + xx.y, rv.z + xx.z, rv.w + xx.w);
    float4 npr = make_float4(sd.x * pr.x + pi.x * inp.x, sd.y * pr.y + pi.y * inp.y,
                             sd.z * pr.z + pi.z * inp.z, sd.w * pr.w + pi.w * inp.w);
    float4 npc = make_float4(sd.x * pc.x + sd.x * pr.x, sd.y * pc.y + sd.y * pr.y,
                             sd.z * pc.z + sd.z * pr.z, sd.w * pc.w + sd.w * pr.w);
    *(float4*)&psc_rise[base4] = npr;
    *(float4*)&psc[base4] = npc;
    float cur = ((npc.x + npc.y) + npc.z) + npc.w;  // left-fold like jnp.sum

    float2 amps = *(const float2*)&asc_amps[n * 2];
    float a1v = asc1[idx], a2v = asc2[idx];
    float na1 = exp_k1[n] * a1v + prev_z * amps.x;
    float na2 = exp_k2[n] * a2v + prev_z * amps.y;
    float vth = v_th[n], el = e_l[n];
    float c1 = cur + a1v + a2v + param_g[n] * el;  // c1 uses OLD asc state
    float nv = decay[n] * v[idx] + current_factor[n] * c1
             + prev_z * (v_reset[n] - vth);
    float v_sc = (nv - vth) / (vth - el);
    z = (v_sc > 0.0f) ? 1.0f : 0.0f;
    float rv0 = rref[idx];
    if (rv0 > 0.0f) z = 0.0f;
    float tr = t_ref[n];
    float nrf = rv0 + z * tr - 1.0f;               // DT = 1
    nrf = fminf(fmaxf(nrf, 0.0f), tr);

    rref[idx] = nrf;
    v[idx] = nv;
    asc1[idx] = na1;
    asc2[idx] = na2;
    out[(t * BB + b) * NN + n] = z;
  }

  // wave32 ballot: pack 32 consecutive neurons' spikes into one ring word.
  uint32_t mask = __builtin_amdgcn_ballot_w32(z > 0.5f);
  if ((threadIdx.x & 31) == 0 && n < NN) {
    zbits[((uint32_t)wslot * NWRD + ((uint32_t)n >> 5)) * 4 + b] = mask;
  }
}

// ---------------------------------------------------------------------------
// Init: zero all state, v = e_l, clear bit ring and rec accumulator.
// ---------------------------------------------------------------------------
__global__ void glif_init(float* __restrict__ rec, float* __restrict__ psc_rise,
                          float* __restrict__ psc, float* __restrict__ v,
                          float* __restrict__ rref, float* __restrict__ asc1,
                          float* __restrict__ asc2, uint32_t* __restrict__ zbits,
                          const float* __restrict__ e_l) {
  int i = blockIdx.x * blockDim.x + threadIdx.x;
  if (i < BB * NRR) { rec[i] = 0.f; psc_rise[i] = 0.f; psc[i] = 0.f; }
  if (i < BB * NN)  { v[i] = e_l[i % NN]; rref[i] = 0.f; asc1[i] = 0.f; asc2[i] = 0.f; }
  if (i < ZB_U32)   zbits[i] = 0u;
}

// ---------------------------------------------------------------------------
// Host driver: carve state out of d_ws, init, then T sequential (edges ->
// neurons) rounds, all on `stream` (graph-capture safe, no allocs/syncs).
// ---------------------------------------------------------------------------
extern "C" void kernel_launch(void* const* d_in, const int* in_sizes, int n_in,
                              void* d_out, int out_size, void* d_ws, size_t ws_size,
                              hipStream_t stream) {
  const float* x              = (const float*)d_in[0];
  const float* weights        = (const float*)d_in[1];
  const float* syn_decay      = (const float*)d_in[2];
  const float* psc_initial    = (const float*)d_in[3];
  const float* asc_amps       = (const float*)d_in[4];
  const float* exp_k1         = (const float*)d_in[5];
  const float* exp_k2         = (const float*)d_in[6];
  const float* decay          = (const float*)d_in[7];
  const float* current_factor = (const float*)d_in[8];
  const float* v_th           = (const float*)d_in[9];
  const float* e_l            = (const float*)d_in[10];
  const float* v_reset        = (const float*)d_in[11];
  const float* t_ref          = (const float*)d_in[12];
  const float* param_g        = (const float*)d_in[13];
  // d_in[14] = dampening (backward pass only; unused in forward)
  const int* rows             = (const int*)d_in[15];
  const int* cols             = (const int*)d_in[16];
  float* out = (float*)d_out;

  // Workspace carve-up (~13 MB total, 256B aligned), L2-resident state.
  char* ws = (char*)d_ws;
  size_t off = 0;
  auto carve = [&](size_t bytes) -> void* {
    void* p = ws + off;
    off = (off + bytes + 255) & ~(size_t)255;
    return p;
  };
  float*    rec      = (float*)carve((size_t)BB * NRR * sizeof(float));
  float*    psc_rise = (float*)carve((size_t)BB * NRR * sizeof(float));
  float*    psc      = (float*)carve((size_t)BB * NRR * sizeof(float));
  float*    v        = (float*)carve((size_t)BB * NN * sizeof(float));
  float*    rref     = (float*)carve((size_t)BB * NN * sizeof(float));
  float*    asc1     = (float*)carve((size_t)BB * NN * sizeof(float));
  float*    asc2     = (float*)carve((size_t)BB * NN * sizeof(float));
  uint32_t* zbits    = (uint32_t*)carve((size_t)ZB_U32 * sizeof(uint32_t));
  (void)ws_size; (void)in_sizes; (void)n_in; (void)out_size;

  {
    int total = BB * NRR;  // dominates all other init extents
    glif_init<<<(total + 255) / 256, 256, 0, stream>>>(
        rec, psc_rise, psc, v, rref, asc1, asc2, zbits, e_l);
  }

  dim3 egrid(EDGE_BLOCKS);          // one-wave blocks, ~2 pipelined tiles each
  dim3 ngrid((NN + 255) / 256, BB); // 196 x 4

  for (int t = 0; t < TT; ++t) {
    glif_edges<<<egrid, 32, 0, stream>>>(rows, cols, weights, zbits, rec, t);
    glif_neurons<<<ngrid, 256, 0, stream>>>(
        x, syn_decay, psc_initial, asc_amps, exp_k1, exp_k2, decay,
        current_factor, v_th, e_l, v_reset, t_ref, param_g,
        rec, psc_rise, psc, v, rref, asc1, asc2, zbits, out, t);
  }
}